// COINs_13786845020209
// MI455X (gfx1250) — compile-verified
//
#include <hip/hip_runtime.h>
#include <hip/hip_bf16.h>

// Problem constants (match reference setup_inputs exactly)
#define NUM_NODES 1048576
#define NUM_COMM  64
#define COMM_SIZE (NUM_NODES / NUM_COMM)   // 16384
#define DD        128
#define EE        262144
#define RR        16
#define TT        8
#define N_INTER   131072

typedef float v2f __attribute__((ext_vector_type(2)));
typedef float v8f __attribute__((ext_vector_type(8)));

// -----------------------------------------------------------------------------
// Fused kernel: xe = x @ W_type^T via V_WMMA_F32_16X16X4_F32, then gather
// comm/intra/inter embeddings, softmax(w3)-weighted combine, L2 normalize,
// write out[2,E,128].
//
// One wave32 per 16 rows of x (rows interleave src/tgt: row i -> s=i&1, e=i>>1).
// WMMA f32 16x16x4 operand layout (wave32):
//   A (16x4): lane l holds M=l&15, K = 2*(l>>4) + {0,1} in the 2 VGPRs
//   B (4x16): lane l holds N=l&15, K = 2*(l>>4) + {0,1} in the 2 VGPRs
//   C/D (16x16): lane l holds N=l&15, rows M = vgpr + 8*(l>>4)
// -----------------------------------------------------------------------------
__global__ __launch_bounds__(256) void coins_main_kernel(
    const float* __restrict__ x,          // [2E, 8]
    const int*   __restrict__ eidx,       // [2, E]
    const int*   __restrict__ cArr,       // [2, E]
    const int*   __restrict__ intra_map,  // [NUM_NODES]
    const int*   __restrict__ inter_map,  // [NUM_NODES]
    const float* __restrict__ W,          // [128, 8] row-major (W_type)
    const float* __restrict__ comm_node,  // [64, 128]
    const float* __restrict__ intra_flat, // [NUM_NODES, 128]
    const float* __restrict__ inter_node, // [N_INTER+1, 128]
    const float* __restrict__ w3,         // [3]
    float*       __restrict__ out)        // [2, E, 128]
{
    const int  lane = threadIdx.x & 31;
    const long gw   = ((long)blockIdx.x * blockDim.x + threadIdx.x) >> 5;
    const long base = gw * 16;                    // first row of x for this wave
    if (base >= 2L * EE) return;                  // exact division: never taken

    // softmax(w3) computed per-thread (3 scalars, L2-resident)
    const float t0 = w3[0], t1 = w3[1], t2 = w3[2];
    const float tm = fmaxf(t0, fmaxf(t1, t2));
    const float x0 = __expf(t0 - tm), x1 = __expf(t1 - tm), x2 = __expf(t2 - tm);
    const float tinv = 1.0f / (x0 + x1 + x2);
    const float w3s0 = x0 * tinv, w3s1 = x1 * tinv, w3s2 = x2 * tinv;

    const int nn   = lane & 15;       // N / M index within the 16-lane group
    const int half = lane >> 4;       // 0: lanes 0-15, 1: lanes 16-31
    const int kb   = half << 1;       // K base for A/B operand pairs

    // ---- A tiles: 16 rows of x, split into K=0..3 and K=4..7 chunks ----
    const float* xrow = x + (base + nn) * TT;
    const v2f A0 = *(const v2f*)(xrow + kb);       // K = kb, kb+1
    const v2f A1 = *(const v2f*)(xrow + 4 + kb);   // K = 4+kb, 4+kb+1

    // ---- GEMM: 8 N-tiles of 16 columns, 2 WMMA (K=4 each) per tile ----
    v8f acc[8];
    #pragma unroll
    for (int j = 0; j < 8; ++j) {
        const float* wp = W + (16 * j + nn) * TT + kb;  // B[k,n] = W_type[n,k]
        const v2f B0 = *(const v2f*)(wp);
        const v2f B1 = *(const v2f*)(wp + 4);
        v8f c = {};
        c = __builtin_amdgcn_wmma_f32_16x16x4_f32(false, A0, false, B0,
                                                  (short)0, c, false, false);
        c = __builtin_amdgcn_wmma_f32_16x16x4_f32(false, A1, false, B1,
                                                  (short)0, c, false, false);
        acc[j] = c;
    }

    // ---- Per-row metadata: lane (l&15) computes row base+(l&15); both halves
    //      duplicate lanes 0..15 (same cache lines, avoids OOB) ----
    const long rowm   = base + nn;
    const int  em     = (int)(rowm >> 1);
    const int  sm     = (int)(rowm & 1);
    const int  c0m    = cArr[em];
    const int  c1m    = cArr[EE + em];
    const int  csm    = sm ? c1m : c0m;           // c[s,e]
    const int  nodem  = eidx[sm * EE + em];       // edge_index[s,e]
    const int  intram = (c0m == c1m) ? 1 : 0;
    const int  idxm   = intram ? (csm * COMM_SIZE + intra_map[nodem])
                               : inter_map[nodem];
    // Prefetch the embedding row we'll gather below (global_prefetch_b8)
    const float* prow = (intram ? intra_flat : inter_node) + (size_t)idxm * DD;
    __builtin_prefetch(prow, 0, 0);
    __builtin_prefetch(prow + 64, 0, 0);

    // ---- Combine + normalize + store, one C/D VGPR slice (row) at a time ----
    #pragma unroll
    for (int v = 0; v < 8; ++v) {
        const int src   = v + (half << 3);        // metadata source lane (0..15)
        const int idx_v = __shfl(idxm,   src, 32);
        const int flg_v = __shfl(intram, src, 32);
        const int cmm_v = __shfl(csm,    src, 32);

        const float* trow = (flg_v ? intra_flat : inter_node) + (size_t)idx_v * DD;
        const float* crow = comm_node + (size_t)cmm_v * DD;

        float ov[8];
        float ss = 0.0f;
        #pragma unroll
        for (int j = 0; j < 8; ++j) {
            const int d = 16 * j + nn;
            const float xv = acc[j][v];           // x_emb contribution (WMMA)
            const float cv = crow[d];             // comm_node gather (L2 hit)
            const float ev = trow[d];             // intra/inter gather
            const float o  = xv * w3s0 + cv * w3s1 + ev * w3s2;
            ov[j] = o;
            ss += o * o;
        }
        // L2-norm reduction across the 16 lanes holding this row
        ss += __shfl_xor(ss, 1, 16);
        ss += __shfl_xor(ss, 2, 16);
        ss += __shfl_xor(ss, 4, 16);
        ss += __shfl_xor(ss, 8, 16);
        const float scale = 1.0f / fmaxf(sqrtf(ss), 1e-12f);

        const long gi = base + v + (half << 3);   // global row of xe
        const int  eo = (int)(gi >> 1);
        const int  so = (int)(gi & 1);
        float* orow = out + ((size_t)so * EE + eo) * DD;
        #pragma unroll
        for (int j = 0; j < 8; ++j)
            orow[16 * j + nn] = ov[j] * scale;    // 64B per half-wave store
    }
}

// -----------------------------------------------------------------------------
// attr_out[e,d] = comm_rel[ea,d]*w2s0 + (intra ? intra_rel[c0,ea,d]
//                                              : inter_rel[ea,d])*w2s1
// 32 threads per edge, float4 per thread: fully coalesced 512B rows.
// -----------------------------------------------------------------------------
__global__ __launch_bounds__(256) void coins_attr_kernel(
    const int*   __restrict__ cArr,       // [2, E]
    const int*   __restrict__ edge_attr,  // [E]
    const float* __restrict__ comm_rel,   // [16, 128]
    const float* __restrict__ intra_rel,  // [64, 16, 128]
    const float* __restrict__ inter_rel,  // [16, 128]
    const float* __restrict__ w2,         // [2]
    float*       __restrict__ attr_out)   // [E, 128]
{
    const long gid = (long)blockIdx.x * blockDim.x + threadIdx.x;
    if (gid >= (long)EE * 32) return;
    const int e = (int)(gid >> 5);
    const int d = ((int)gid & 31) * 4;

    const float u0 = w2[0], u1 = w2[1];
    const float um = fmaxf(u0, u1);
    const float y0 = __expf(u0 - um), y1 = __expf(u1 - um);
    const float yinv = 1.0f / (y0 + y1);
    const float w2s0 = y0 * yinv, w2s1 = y1 * yinv;

    const int  ea    = edge_attr[e];
    const int  c0    = cArr[e];
    const int  c1    = cArr[EE + e];
    const bool intra = (c0 == c1);

    const float4 a = *(const float4*)(comm_rel + (size_t)ea * DD + d);
    const float* bp = intra ? (intra_rel + ((size_t)c0 * RR + ea) * DD + d)
                            : (inter_rel + (size_t)ea * DD + d);
    const float4 b = *(const float4*)bp;

    float4 r;
    r.x = a.x * w2s0 + b.x * w2s1;
    r.y = a.y * w2s0 + b.y * w2s1;
    r.z = a.z * w2s0 + b.z * w2s1;
    r.w = a.w * w2s0 + b.w * w2s1;
    *(float4*)(attr_out + (size_t)e * DD + d) = r;
}

extern "C" void kernel_launch(void* const* d_in, const int* in_sizes, int n_in,
                              void* d_out, int out_size, void* d_ws, size_t ws_size,
                              hipStream_t stream) {
    (void)in_sizes; (void)n_in; (void)out_size; (void)d_ws; (void)ws_size;
    // setup_inputs() order:
    const float* x          = (const float*)d_in[0];   // [2E, 8]
    const int*   edge_index = (const int*)  d_in[1];   // [2, E]
    const int*   cArr       = (const int*)  d_in[2];   // [2, E]
    const int*   edge_attr  = (const int*)  d_in[3];   // [E]
    const int*   intra_map  = (const int*)  d_in[4];   // [NUM_NODES]
    const int*   inter_map  = (const int*)  d_in[5];   // [NUM_NODES]
    const float* W_type     = (const float*)d_in[6];   // [128, 8]
    const float* comm_node  = (const float*)d_in[7];   // [64, 128]
    const float* comm_rel   = (const float*)d_in[8];   // [16, 128]
    const float* intra_flat = (const float*)d_in[9];   // [NUM_NODES, 128]
    const float* intra_rel  = (const float*)d_in[10];  // [64, 16, 128]
    const float* inter_node = (const float*)d_in[11];  // [N_INTER+1, 128]
    const float* inter_rel  = (const float*)d_in[12];  // [16, 128]
    const float* w3         = (const float*)d_in[13];  // [3]
    const float* w2         = (const float*)d_in[14];  // [2]

    float* out      = (float*)d_out;                   // [2, E, 128]
    float* attr_out = out + (size_t)2 * EE * DD;       // [E, 128]

    // Main fused kernel: one wave32 per 16 rows of x -> 32768 waves.
    {
        const int threads = 256;                       // 8 waves per block
        const int waves   = (2 * EE) / 16;             // 32768
        const int blocks  = waves / (threads / 32);    // 4096
        coins_main_kernel<<<blocks, threads, 0, stream>>>(
            x, edge_index, cArr, intra_map, inter_map, W_type,
            comm_node, intra_flat, inter_node, w3, out);
    }
    // Attr kernel: 32 threads x float4 per edge.
    {
        const int threads = 256;
        const long total  = (long)EE * 32;
        const int blocks  = (int)((total + threads - 1) / threads);  // 32768
        coins_attr_kernel<<<blocks, threads, 0, stream>>>(
            cArr, edge_attr, comm_rel, intra_rel, inter_rel, w2, attr_out);
    }
}